// LGAE_IP_Linear_40699110097056
// MI455X (gfx1250) — compile-verified
//
#include <hip/hip_runtime.h>

typedef __attribute__((ext_vector_type(2))) float v2f;
typedef __attribute__((ext_vector_type(8))) float v8f;

#define NCOLS 256           // fixed N dimension (HID) for both GEMMs
#define KC    32            // K-chunk staged in LDS
#define BPAIR 544           // floats per interleaved (even,odd) K-row pair: 512+32
#define AROW  36            // floats per A row in LDS (conflict-free 16x4 fetch)
#define APLANE (128 * AROW)      // floats per A buffer plane
#define BPLANE ((KC / 2) * BPAIR) // floats per B buffer plane

// ---------------------------------------------------------------------------
// C[M x 256] = A[M x K] @ B[K x 256], all f32, V_WMMA_F32_16X16X4_F32.
// grid.x = M/128, block = 256 threads = 8 wave32.
// Double-buffered LDS pipeline:
//   A chunk (128 x KC): global_load_async_to_lds_b128, th:TH_LOAD_NT
//     (adj = 1 GB single-use stream; NT keeps it out of L2 so H/Z stay hot).
//   B chunk (KC x 256): cached loads, even/odd K rows interleaved in LDS so
//     each WMMA B operand is one paired ds_load_b64.
//   Chunk c+1's asyncs issue before waiting asynccnt<=4 for chunk c (async
//   loads retire in order per wave); B staging overlaps the compute region.
// ---------------------------------------------------------------------------
__global__ __launch_bounds__(256) void gemm256_f32_wmma(
    const float* __restrict__ A, const float* __restrict__ B,
    float* __restrict__ C, int M, int K)
{
  __shared__ float ldsA[2 * APLANE];   // 2 x 18 KB
  __shared__ float ldsB[2 * BPLANE];   // 2 x 34 KB

  const int tid   = threadIdx.x;
  const int wave  = tid >> 5;
  const int lane  = tid & 31;
  const int laneN = lane & 15;
  const int half  = lane >> 4;

  const int m0 = blockIdx.x * 128;

  // --- async A staging mapping: (row = tid>>3 + 32*i, float4 col = tid&7)
  const int ar  = tid >> 3;
  const int ac4 = tid & 7;
  const unsigned ldsA_base = (unsigned)(size_t)(&ldsA[0]);
  const unsigned a_lds0 = ldsA_base + (unsigned)(ar * AROW * 4 + ac4 * 16);
  const unsigned a_va0  = (unsigned)(((size_t)ar * (size_t)K + ac4 * 4) * 4);

  // --- B staging mapping: 8 iters, (row-pair, col-pair) = (i*2 + t>>7, t&127)
  const int brp0 = tid >> 7;
  const int bcp  = tid & 127;

  // --- compute-side LDS addressing
  const int aoff0 = (wave * 16 + laneN) * AROW + 2 * half;
  const int bbase0 = half * BPAIR + laneN * 2;

  // stage chunk kc's A rows into LDS plane `buf` (4 async b128 per thread)
  auto stageA = [&](int kc, int buf) {
    const float* Ab = A + (size_t)m0 * (size_t)K + kc;   // uniform -> SGPR pair
    const unsigned lb = a_lds0 + (unsigned)(buf * (int)(APLANE * 4));
    #pragma unroll
    for (int i = 0; i < 4; ++i) {
      const unsigned lo = lb + (unsigned)(i * 32 * AROW * 4);
      const unsigned va = a_va0 + (unsigned)(i * 32 * (size_t)K * 4);
      asm volatile("global_load_async_to_lds_b128 %0, %1, %2 th:TH_LOAD_NT"
                   :: "v"(lo), "v"(va), "s"(Ab)
                   : "memory");
    }
  };

  // stage chunk kc's B rows (interleaved even/odd) into LDS plane `buf`
  auto stageB = [&](int kc, int buf) {
    float* dst = &ldsB[buf * BPLANE];
    #pragma unroll
    for (int i = 0; i < 8; ++i) {
      const int rp = i * 2 + brp0;
      const float2 e = *(const float2*)(B + (size_t)(kc + rp * 2)     * NCOLS + bcp * 2);
      const float2 o = *(const float2*)(B + (size_t)(kc + rp * 2 + 1) * NCOLS + bcp * 2);
      const float4 v = make_float4(e.x, o.x, e.y, o.y);
      *(float4*)(&dst[rp * BPAIR + bcp * 4]) = v;
    }
  };

  v8f acc[16];
  #pragma unroll
  for (int t = 0; t < 16; ++t) {
    v8f z = {0.f, 0.f, 0.f, 0.f, 0.f, 0.f, 0.f, 0.f};
    acc[t] = z;
  }

  // prologue: chunk 0 into plane 0
  stageA(0, 0);
  stageB(0, 0);

  int buf = 0;
  for (int kc = 0; kc < K; kc += KC, buf ^= 1) {
    const bool has_next = (kc + KC) < K;
    if (has_next) {
      stageA(kc + KC, buf ^ 1);
      asm volatile("s_wait_asynccnt 4" ::: "memory");  // chunk kc landed
    } else {
      asm volatile("s_wait_asynccnt 0" ::: "memory");
    }
    __syncthreads();   // barrier1: chunk kc (A async + B stores) visible

    if (has_next) stageB(kc + KC, buf ^ 1);  // overlaps compute below

    const float* aop = &ldsA[buf * APLANE + aoff0];
    const float* bop = &ldsB[buf * BPLANE];
    #pragma unroll
    for (int kk = 0; kk < KC; kk += 4) {
      const v2f a = *(const v2f*)(aop + kk);                 // 16x4 A operand
      const int bb = (kk >> 1) * BPAIR + bbase0;
      #pragma unroll
      for (int nt = 0; nt < 16; ++nt) {
        const v2f b = *(const v2f*)(&bop[bb + nt * 32]);     // paired 4x16 B
        acc[nt] = __builtin_amdgcn_wmma_f32_16x16x4_f32(
            false, a, false, b, (short)0, acc[nt], false, false);
      }
    }
    __syncthreads();   // barrier2: plane `buf` free for re-staging
  }

  // D layout (16x16 f32): VGPR r -> (M = r + 8*half, N = laneN) per tile.
  const int mrow = m0 + wave * 16;
  #pragma unroll
  for (int nt = 0; nt < 16; ++nt) {
    #pragma unroll
    for (int r = 0; r < 8; ++r) {
      C[(size_t)(mrow + r + 8 * half) * NCOLS + nt * 16 + laneN] = acc[nt][r];
    }
  }
}

// ---------------------------------------------------------------------------
// Edge scoring: out[e] = sigmoid( sum_k Z[i][k] * Z[j][k] * w2[k] )
// One wave per edge (grid-stride). Lane l owns columns [8l, 8l+8).
// Z (16 MB) is L2-resident; w2 lives in registers across the edge loop.
// ---------------------------------------------------------------------------
__global__ __launch_bounds__(256) void edge_score_kernel(
    const float* __restrict__ Z, const float* __restrict__ w2,
    const int* __restrict__ edges_true, const int* __restrict__ edges_false,
    float* __restrict__ out, int E)
{
  const int lane   = threadIdx.x & 31;
  const int gwave  = (int)((blockIdx.x * blockDim.x + threadIdx.x) >> 5);
  const int nwaves = (int)((gridDim.x * blockDim.x) >> 5);

  const float4 w0 = *(const float4*)(w2 + lane * 8);
  const float4 w1 = *(const float4*)(w2 + lane * 8 + 4);

  const int total = 2 * E;
  for (int e = gwave; e < total; e += nwaves) {
    int i, j;
    if (e < E) {
      i = edges_true[2 * e];
      j = edges_true[2 * e + 1];
    } else {
      const int f = e - E;
      i = edges_false[2 * f];
      j = edges_false[2 * f + 1];
    }
    const float* zi = Z + (size_t)i * NCOLS + lane * 8;
    const float* zj = Z + (size_t)j * NCOLS + lane * 8;
    const float4 a0 = *(const float4*)zi;
    const float4 a1 = *(const float4*)(zi + 4);
    const float4 b0 = *(const float4*)zj;
    const float4 b1 = *(const float4*)(zj + 4);

    float p = a0.x * b0.x * w0.x + a0.y * b0.y * w0.y +
              a0.z * b0.z * w0.z + a0.w * b0.w * w0.w +
              a1.x * b1.x * w1.x + a1.y * b1.y * w1.y +
              a1.z * b1.z * w1.z + a1.w * b1.w * w1.w;

    #pragma unroll
    for (int off = 16; off >= 1; off >>= 1)
      p += __shfl_xor(p, off, 32);

    if (lane == 0) out[e] = 1.0f / (1.0f + __expf(-p));
  }
}

// ---------------------------------------------------------------------------
extern "C" void kernel_launch(void* const* d_in, const int* in_sizes, int n_in,
                              void* d_out, int out_size, void* d_ws, size_t ws_size,
                              hipStream_t stream) {
  const float* X   = (const float*)d_in[0];   // (16384, 512)
  const float* adj = (const float*)d_in[1];   // (16384, 16384)
  const float* W   = (const float*)d_in[2];   // (512, 256)
  const float* w2  = (const float*)d_in[3];   // (256, 1)
  const int*   e1  = (const int*)d_in[4];     // (E, 2)
  const int*   e2  = (const int*)d_in[5];     // (E, 2)
  float* out = (float*)d_out;                 // (2E, 1)

  const int N  = 16384;
  const int K1 = 512;
  const int E  = in_sizes[4] / 2;

  float* H = (float*)d_ws;                    // 16384*256 f32 = 16 MB
  float* Z = H + (size_t)N * NCOLS;           // 16384*256 f32 = 16 MB

  // H = X @ W          (X = 32 MB streamed once, NT)
  gemm256_f32_wmma<<<N / 128, 256, 0, stream>>>(X, W, H, N, K1);
  // Z = adj @ H        (adj = 1 GB streamed once, NT -> ~43 us HBM floor)
  gemm256_f32_wmma<<<N / 128, 256, 0, stream>>>(adj, H, Z, N, N);
  // logits + sigmoid   (Z gathers hit L2)
  edge_score_kernel<<<2048, 256, 0, stream>>>(Z, w2, e1, e2, out, E);
}